// CAMD_9990093930844
// MI455X (gfx1250) — compile-verified
//
#include <hip/hip_runtime.h>

// Problem constants (from reference)
#define T_LEN 4096
#define D_IN  66          // feature dim incl. timestamp
#define D_PAD 80          // padded N dim: 5 tiles of 16
#define K_PAD 68          // padded K dim: 17 steps of 4
#define N_LAYERS 3
#define ROWS_PER_WG 64    // 4 waves x 16 rows

typedef float v2f __attribute__((ext_vector_type(2)));
typedef float v8f __attribute__((ext_vector_type(8)));

// ---------------------------------------------------------------------------
// Kernel 1: 5 MLPs (Q from x1 with wq/bq; K_m from x_m with wk/bk).
// Each wave computes one 16-row tile; WMMA f32 16x16x4 for every matmul.
// ---------------------------------------------------------------------------
__global__ __launch_bounds__(128)
void camd_mlp_kernel(const float* __restrict__ x1, const float* __restrict__ x2,
                     const float* __restrict__ x3, const float* __restrict__ x4,
                     const float* __restrict__ wq, const float* __restrict__ bq,
                     const float* __restrict__ wk, const float* __restrict__ bk,
                     float* __restrict__ qpad, float* __restrict__ kpad)
{
    __shared__ float sW[K_PAD * D_PAD];            // one layer's weights, zero-padded
    __shared__ float sB[D_PAD];                    // bias, zero-padded
    __shared__ float actA[ROWS_PER_WG * D_PAD];    // ping
    __shared__ float actB[ROWS_PER_WG * D_PAD];    // pong

    const int tid  = threadIdx.x;
    const int wave = tid >> 5;
    const int lane = tid & 31;
    const int p    = blockIdx.y;                   // 0 => Q-MLP, 1..4 => K-MLP on x_p

    const float* xin = (p <= 1) ? x1 : (p == 2 ? x2 : (p == 3 ? x3 : x4));
    const float* Wg  = (p == 0) ? wq : wk;
    const float* Bg  = (p == 0) ? bq : bk;
    float* gout = (p == 0) ? qpad : (kpad + (size_t)(p - 1) * T_LEN * D_PAD);

    const int rowBase = blockIdx.x * ROWS_PER_WG;

    // Load input tile into actA, zero-padding cols >= 66.
    for (int idx = tid; idx < ROWS_PER_WG * D_PAD; idx += blockDim.x) {
        int r = idx / D_PAD, c = idx % D_PAD;
        actA[idx] = (c < D_IN) ? xin[(size_t)(rowBase + r) * D_IN + c] : 0.f;
    }

    const int ml = lane & 15;      // row within 16-tile (A) / col within N-tile (B,C)
    const int hs = lane >> 4;      // half-select: K pairs (A,B), row+8 (C/D)

    float* bufIn  = actA;
    float* bufOut = actB;

    for (int layer = 0; layer < N_LAYERS; ++layer) {
        __syncthreads();   // previous compute done before overwriting sW / reading acts
        for (int idx = tid; idx < K_PAD * D_PAD; idx += blockDim.x) {
            int k = idx / D_PAD, n = idx % D_PAD;
            sW[idx] = (k < D_IN && n < D_IN)
                        ? Wg[(size_t)layer * D_IN * D_IN + (size_t)k * D_IN + n] : 0.f;
        }
        if (tid < D_PAD) sB[tid] = (tid < D_IN) ? Bg[layer * D_IN + tid] : 0.f;
        __syncthreads();

        const float* aRow = bufIn + (wave * 16 + ml) * D_PAD;   // this lane's A row
        for (int nt = 0; nt < 5; ++nt) {
            const int n = nt * 16 + ml;
            v8f acc = {};
            for (int ks = 0; ks < 17; ++ks) {
                const int kk = ks * 4 + 2 * hs;                 // A/B K-pair per half
                v2f a = *(const v2f*)(aRow + kk);               // A[m=ml][kk..kk+1]
                v2f bf;
                bf.x = sW[kk * D_PAD + n];                      // B[kk][n]
                bf.y = sW[(kk + 1) * D_PAD + n];                // B[kk+1][n]
                acc = __builtin_amdgcn_wmma_f32_16x16x4_f32(
                        /*neg_a=*/false, a, /*neg_b=*/false, bf,
                        /*c_mod=*/(short)0, acc,
                        /*reuse_a=*/false, /*reuse_b=*/false);
            }
            const float bias = sB[n];
            for (int r = 0; r < 8; ++r) {
                const int row = r + 8 * hs;                     // C/D layout
                float v = acc[r] + bias;
                if (layer < N_LAYERS - 1) {
                    v = fmaxf(v, 0.f);
                    bufOut[(wave * 16 + row) * D_PAD + n] = v;
                } else {
                    gout[(size_t)(rowBase + wave * 16 + row) * D_PAD + n] = v;
                }
            }
        }
        float* t = bufIn; bufIn = bufOut; bufOut = t;
    }
}

// ---------------------------------------------------------------------------
// Kernel 2: cumulative state.  cum[m][t][d*2+v] = sum_{j<=t} K_m[j,d]*x_m[j,v].
// One workgroup per (m,d,v) series: 16 sequential chunks of a 256-wide LDS scan.
// ---------------------------------------------------------------------------
__global__ __launch_bounds__(256)
void camd_scan_kernel(const float* __restrict__ x1, const float* __restrict__ x2,
                      const float* __restrict__ x3, const float* __restrict__ x4,
                      const float* __restrict__ kpad, float* __restrict__ cum)
{
    __shared__ float sbuf[256];
    const int tid = threadIdx.x;
    const int bid = blockIdx.x;
    const int m = bid / (D_IN * 2);
    const int s = bid % (D_IN * 2);
    const int d = s >> 1;
    const int v = s & 1;
    const float* xm = (m == 0) ? x1 : (m == 1) ? x2 : (m == 2) ? x3 : x4;
    const float* kb = kpad + (size_t)m * T_LEN * D_PAD;
    float* cm = cum + (size_t)m * T_LEN * (D_IN * 2);

    float carry = 0.f;
    for (int chunk = 0; chunk < T_LEN / 256; ++chunk) {
        const int t = chunk * 256 + tid;
        sbuf[tid] = kb[(size_t)t * D_PAD + d] * xm[(size_t)t * D_IN + v];
        __syncthreads();
        for (int off = 1; off < 256; off <<= 1) {
            float add = (tid >= off) ? sbuf[tid - off] : 0.f;
            __syncthreads();
            sbuf[tid] += add;
            __syncthreads();
        }
        cm[(size_t)t * (D_IN * 2) + s] = carry + sbuf[tid];
        carry += sbuf[255];
        __syncthreads();                 // protect sbuf[255] read vs next chunk
    }
}

// ---------------------------------------------------------------------------
// Kernel 3: per-query binary search (searchsorted right - 1) + gather-dot.
// One wave32 per query; dims split d = lane, lane+32, (64,65 on lanes 0,1).
// ---------------------------------------------------------------------------
__global__ __launch_bounds__(128)
void camd_attn_kernel(const float* __restrict__ x1, const float* __restrict__ x2,
                      const float* __restrict__ x3, const float* __restrict__ x4,
                      const float* __restrict__ qpad, const float* __restrict__ cum,
                      float* __restrict__ out)
{
    const int tid  = threadIdx.x;
    const int wave = tid >> 5;
    const int lane = tid & 31;
    const int i = blockIdx.x * 4 + wave;

    const float q0 = qpad[(size_t)i * D_PAD + lane];
    const float q1 = qpad[(size_t)i * D_PAD + 32 + lane];
    const float q2 = (lane < 2) ? qpad[(size_t)i * D_PAD + 64 + lane] : 0.f;
    const float t1v = x1[(size_t)i * D_IN + (D_IN - 1)];

    float acc0 = 0.f, acc1 = 0.f;
    for (int m = 0; m < 4; ++m) {
        const float* xm = (m == 0) ? x1 : (m == 1) ? x2 : (m == 2) ? x3 : x4;
        // count of { j : t2[j] <= t1v }  (timestamps sorted, stride D_IN)
        int lo = 0, hi = T_LEN;
        while (lo < hi) {
            const int mid = (lo + hi) >> 1;
            const float t2 = xm[(size_t)mid * D_IN + (D_IN - 1)];
            if (t2 <= t1v) lo = mid + 1; else hi = mid;
        }
        const int idx = lo - 1;
        if (idx >= 0) {
            const float* row = cum + ((size_t)m * T_LEN + idx) * (D_IN * 2);
            acc0 += q0 * row[lane * 2 + 0] + q1 * row[(lane + 32) * 2 + 0];
            acc1 += q0 * row[lane * 2 + 1] + q1 * row[(lane + 32) * 2 + 1];
            if (lane < 2) {
                acc0 += q2 * row[(lane + 64) * 2 + 0];
                acc1 += q2 * row[(lane + 64) * 2 + 1];
            }
        }
    }
    for (int off = 16; off > 0; off >>= 1) {
        acc0 += __shfl_xor(acc0, off, 32);
        acc1 += __shfl_xor(acc1, off, 32);
    }
    if (lane == 0) {
        out[(size_t)i * 2 + 0] = acc0;
        out[(size_t)i * 2 + 1] = acc1;
    }
}

// ---------------------------------------------------------------------------
extern "C" void kernel_launch(void* const* d_in, const int* in_sizes, int n_in,
                              void* d_out, int out_size, void* d_ws, size_t ws_size,
                              hipStream_t stream) {
    (void)in_sizes; (void)n_in; (void)out_size; (void)ws_size;
    const float* x1 = (const float*)d_in[0];
    const float* x2 = (const float*)d_in[1];
    const float* x3 = (const float*)d_in[2];
    const float* x4 = (const float*)d_in[3];
    const float* wq = (const float*)d_in[4];
    const float* bq = (const float*)d_in[5];
    const float* wk = (const float*)d_in[6];
    const float* bk = (const float*)d_in[7];
    float* out = (float*)d_out;

    float* ws   = (float*)d_ws;
    float* qpad = ws;                                    // 4096*80
    float* kpad = qpad + (size_t)T_LEN * D_PAD;          // 4*4096*80
    float* cum  = kpad + (size_t)4 * T_LEN * D_PAD;      // 4*4096*132

    camd_mlp_kernel<<<dim3(T_LEN / ROWS_PER_WG, 5), 128, 0, stream>>>(
        x1, x2, x3, x4, wq, bq, wk, bk, qpad, kpad);
    camd_scan_kernel<<<4 * D_IN * 2, 256, 0, stream>>>(
        x1, x2, x3, x4, kpad, cum);
    camd_attn_kernel<<<T_LEN / 4, 128, 0, stream>>>(
        x1, x2, x3, x4, qpad, cum, out);
}